// ORAlign2d_80401787781762
// MI455X (gfx1250) — compile-verified
//
#include <hip/hip_runtime.h>
#include <stdint.h>

// ORAlign2d (RIE feature alignment), specialized for nOrientation = 8.
// Input:  [N, C, 1, 1] float32 with C = nFeature * 8 (N=4096, C=16384).
// Per group of 8 contiguous floats: find first argmax m, output rotate-left by m.
//
// Pure streaming kernel: 512 MB total traffic -> ~22 us floor at 23.3 TB/s.
// MI455X-specific path: inbound stream goes through the CDNA5 async
// global->LDS copy engine (GLOBAL_LOAD_ASYNC_TO_LDS_B128 / ASYNCcnt), each
// lane staging its own 32-byte group into a private LDS slot (no barrier
// needed, only s_wait_asynccnt). Output uses non-temporal B128 stores so the
// write stream does not evict useful L2 lines.

#define TPB 256  // 8 wave32s per block

// True vector type (HIP's float4 is a struct and is rejected by
// __builtin_nontemporal_store).
typedef float v4f __attribute__((ext_vector_type(4)));

__global__ __launch_bounds__(TPB) void oralign8_kernel(
    const float* __restrict__ in, float* __restrict__ out, int nGroups)
{
    __shared__ float stage[TPB * 8];  // 8 KB: 32 bytes per thread

    const int g = blockIdx.x * TPB + threadIdx.x;

    // 32-bit LDS byte address of this thread's slot (low 32 bits of the
    // generic pointer are the LDS offset within the shared aperture).
    unsigned lds_addr = (unsigned)(uintptr_t)(&stage[threadIdx.x * 8]);

    if (g < nGroups) {
        // 64-bit global address of this group's 32 bytes.
        unsigned long long gaddr =
            (unsigned long long)(uintptr_t)(in + (size_t)g * 8);
        // Async copy 32 bytes global -> LDS (two B128 ops). INST_OFFSET is
        // applied to both the global and LDS addresses, so offset:16 lands
        // the second half in the right LDS slot half.
        asm volatile(
            "global_load_async_to_lds_b128 %0, %1, off\n\t"
            "global_load_async_to_lds_b128 %0, %1, off offset:16"
            :
            : "v"(lds_addr), "v"(gaddr)
            : "memory");
    }
    // Drain this wave's async copies; each lane only reads back its own slot,
    // so no workgroup barrier is required.
    asm volatile("s_wait_asynccnt 0" ::: "memory");

    if (g < nGroups) {
        const v4f* sv = (const v4f*)(&stage[threadIdx.x * 8]);
        v4f a = sv[0];   // ds_load_b128
        v4f b = sv[1];   // ds_load_b128

        float x[8] = {a.x, a.y, a.z, a.w, b.x, b.y, b.z, b.w};

        // First-occurrence argmax (matches jnp.argmax tie-break).
        int   m    = 0;
        float best = x[0];
#pragma unroll
        for (int k = 1; k < 8; ++k) {
            if (x[k] > best) { best = x[k]; m = k; }
        }

        // Rotate left by m using a log-shift select network: three fixed
        // rotates (by 1, 2, 4) each gated by one bit of m. All indices are
        // compile-time constants after unrolling -> stays in VGPRs
        // (v_cndmask chains), no scratch.
#pragma unroll
        for (int s = 1; s <= 4; s <<= 1) {
            const bool doit = (m & s) != 0;
            float y[8];
#pragma unroll
            for (int k = 0; k < 8; ++k) y[k] = x[(k + s) & 7];
#pragma unroll
            for (int k = 0; k < 8; ++k) x[k] = doit ? y[k] : x[k];
        }

        // Streamed output: non-temporal 128-bit stores.
        v4f o0 = { x[0], x[1], x[2], x[3] };
        v4f o1 = { x[4], x[5], x[6], x[7] };
        v4f* dst = (v4f*)(out + (size_t)g * 8);
        __builtin_nontemporal_store(o0, dst + 0);
        __builtin_nontemporal_store(o1, dst + 1);
    }
}

extern "C" void kernel_launch(void* const* d_in, const int* in_sizes, int n_in,
                              void* d_out, int out_size, void* d_ws, size_t ws_size,
                              hipStream_t stream) {
    const float* in  = (const float*)d_in[0];
    float*       out = (float*)d_out;
    // d_in[1] is nOrientation (== 8 per the reference setup); kernel is
    // specialized for O = 8.
    (void)n_in; (void)d_ws; (void)ws_size; (void)out_size;

    const int total   = in_sizes[0];       // N * C = 4096 * 16384
    const int nGroups = total / 8;         // 8,388,608 groups
    const int blocks  = (nGroups + TPB - 1) / TPB;

    oralign8_kernel<<<blocks, TPB, 0, stream>>>(in, out, nGroups);
}